// NaiveTemporalShift_71923522339051
// MI455X (gfx1250) — compile-verified
//
#include <hip/hip_runtime.h>

// Temporal shift (B,T,N,C)=(64,128,21,256), U=1 -> partitions [86,85,85],
// shifts [-1,0,+1]. Pure HBM-bandwidth problem: ~352 MB @ 23.3 TB/s ~= 15 us.
// Strategy: one float4 (B128) per lane, non-temporal load/store, gfx1250
// global_prefetch_b8 stream-ahead. Only 2 of 64 channel-groups straddle a
// partition boundary and take a scalar-dword path.

typedef float v4f __attribute__((ext_vector_type(4)));

__global__ __launch_bounds__(256) void temporal_shift_b128_kernel(
    const float* __restrict__ x, float* __restrict__ out, int total4) {
  constexpr int T = 128;
  constexpr int N = 21;
  constexpr int C = 256;
  constexpr int TS = N * C;            // elements per time step (5376)

  int tid = blockIdx.x * blockDim.x + threadIdx.x;
  if (tid >= total4) return;

  int g = tid & 63;                    // float4 group within C (0..63)
  int r = tid >> 6;                    // row index over (b,t,n)
  int t = (r / N) & (T - 1);           // time index (T is a power of two)
  int off = r * C + (g << 2);          // element offset of this float4

  // gfx1250 speculative prefetch: warm L2 ahead of the stream. OOB near the
  // buffer end translates-fail silently (speculative prefetch is dropped).
  __builtin_prefetch(x + off + 8 * TS, /*rw=*/0, /*locality=*/1);

  const v4f zero = {0.f, 0.f, 0.f, 0.f};
  v4f v;

  if (g < 21) {
    // channels 0..83 : shift -1 -> read t+1 (zero at t = T-1)
    v = (t < T - 1) ? __builtin_nontemporal_load((v4f*)(x + off + TS)) : zero;
  } else if (g == 21) {
    // channels 84..87 : c=84,85 shift -1 ; c=86,87 shift 0
    float a0 = 0.f, a1 = 0.f;
    if (t < T - 1) {
      a0 = x[off + TS];
      a1 = x[off + TS + 1];
    }
    v.x = a0;
    v.y = a1;
    v.z = x[off + 2];
    v.w = x[off + 3];
  } else if (g < 42) {
    // channels 88..167 : shift 0 -> straight copy
    v = __builtin_nontemporal_load((v4f*)(x + off));
  } else if (g == 42) {
    // channels 168..171 : c=168..170 shift 0 ; c=171 shift +1
    v.x = x[off];
    v.y = x[off + 1];
    v.z = x[off + 2];
    v.w = (t > 0) ? x[off + 3 - TS] : 0.f;
  } else {
    // channels 172..255 : shift +1 -> read t-1 (zero at t = 0)
    v = (t > 0) ? __builtin_nontemporal_load((v4f*)(x + off - TS)) : zero;
  }

  __builtin_nontemporal_store(v, (v4f*)(out + off));
}

extern "C" void kernel_launch(void* const* d_in, const int* in_sizes, int n_in,
                              void* d_out, int out_size, void* d_ws, size_t ws_size,
                              hipStream_t stream) {
  const float* x = (const float*)d_in[0];
  float* out = (float*)d_out;

  // out_size = B*T*N*C = 44,040,192 floats; C=256 divisible by 4.
  int total4 = out_size >> 2;          // float4 count = 11,010,048
  const int threads = 256;             // 8 waves of 32 on CDNA5
  int blocks = (total4 + threads - 1) / threads;

  temporal_shift_b128_kernel<<<blocks, threads, 0, stream>>>(x, out, total4);
}